// ModelNew_25056839204955
// MI455X (gfx1250) — compile-verified
//
#include <hip/hip_runtime.h>

typedef __attribute__((ext_vector_type(2))) float v2f;
typedef __attribute__((ext_vector_type(8))) float v8f;

#define B_ 32
#define N_ 1024
#define D_ 512
#define K_ 64
#define EPS_ 1e-12f

// ---------------------------------------------------------------------------
// Kernel 1: row softmax over K=64. One wave32 per row, 2 elements per lane.
// ---------------------------------------------------------------------------
__global__ void __launch_bounds__(256) softmax_k(const float* __restrict__ bn,
                                                 float* __restrict__ P) {
    int wave = threadIdx.x >> 5;
    int lane = threadIdx.x & 31;
    int row  = blockIdx.x * 8 + wave;            // 4096 blocks * 8 = 32768 rows
    const float* src = bn + (size_t)row * K_;
    float a = src[lane];
    float b = src[lane + 32];
    float m = fmaxf(a, b);
    #pragma unroll
    for (int off = 16; off > 0; off >>= 1) m = fmaxf(m, __shfl_xor(m, off, 32));
    float e0 = __expf(a - m);
    float e1 = __expf(b - m);
    float s = e0 + e1;
    #pragma unroll
    for (int off = 16; off > 0; off >>= 1) s += __shfl_xor(s, off, 32);
    float inv = 1.0f / s;
    float* dst = P + (size_t)row * K_;
    dst[lane]      = e0 * inv;
    dst[lane + 32] = e1 * inv;
}

// ---------------------------------------------------------------------------
// Kernel 2: a_sum[b,k] = sum_n P[b,n,k]. Deterministic tree reduce.
// ---------------------------------------------------------------------------
__global__ void __launch_bounds__(256) asum_k(const float* __restrict__ P,
                                              float* __restrict__ a_sum) {
    __shared__ float red[256];
    int b = blockIdx.x;
    int k = threadIdx.x & 63;
    int g = threadIdx.x >> 6;
    const float* Pb = P + (size_t)b * N_ * K_;
    float s = 0.0f;
    for (int n = g; n < N_; n += 4) s += Pb[(size_t)n * K_ + k];
    red[threadIdx.x] = s;
    __syncthreads();
    if (g == 0)
        a_sum[b * K_ + k] = red[k] + red[64 + k] + red[128 + k] + red[192 + k];
}

// ---------------------------------------------------------------------------
// Issue one per-lane 16-byte async global->LDS copy (ASYNCcnt-tracked).
// ---------------------------------------------------------------------------
__device__ __forceinline__ void async_b128(void* lds_ptr, const void* gptr) {
    unsigned lds = (unsigned)(uintptr_t)lds_ptr;   // generic shared ptr: low 32 bits = LDS offset
    unsigned long long ga = (unsigned long long)(uintptr_t)gptr;
    asm volatile("global_load_async_to_lds_b128 %0, %1, off"
                 :: "v"(lds), "v"(ga) : "memory");
}

__device__ __forceinline__ void wait_async0() {
    asm volatile("s_wait_asynccnt 0x0" ::: "memory");
}

// ---------------------------------------------------------------------------
// Kernel 3: vlad[b,k,d] = sum_n P[b,n,k] * x[b,n,d] via V_WMMA_F32_16X16X4_F32.
// Grid: B * (D/64) = 256 blocks, 256 threads (8 waves).
// Each wave owns a 16(k) x 32(d) output (2 accumulators, shared A fragment).
// Double-buffered LDS fed by async global->LDS copies; copy of chunk c+1
// overlaps WMMA of chunk c. One barrier per chunk.
// ---------------------------------------------------------------------------
__global__ void __launch_bounds__(256) vlad_gemm_k(const float* __restrict__ P,
                                                   const float* __restrict__ x,
                                                   float* __restrict__ vlad) {
    __shared__ __align__(16) float sP[2][32 * 64];   // [buf][nn][k]  2 x 8 KB
    __shared__ __align__(16) float sX[2][32 * 64];   // [buf][nn][dd] 2 x 8 KB

    int b      = blockIdx.x >> 3;          // 8 d-blocks of 64
    int d0base = (blockIdx.x & 7) * 64;
    int t      = threadIdx.x;
    int wave   = t >> 5;
    int lane   = t & 31;
    int k0     = (wave & 3) * 16;          // 4 k-tiles
    int dg     = (wave >> 2) * 32;         // 2 d-groups of 32
    int half   = lane >> 4;                // 0: K=0,1   1: K=2,3
    int l      = lane & 15;

    const float* Pb = P + (size_t)b * N_ * K_;
    const float* xb = x + (size_t)b * N_ * D_;

    // Staging (per thread, float4 granularity). Both tiles are 512 float4.
    int f0 = t, f1 = t + 256;
    int xr0 = f0 >> 4, xc0 = (f0 & 15) * 4;   // X tile: row = f>>4, col = (f&15)*4
    int xr1 = f1 >> 4, xc1 = (f1 & 15) * 4;

    v8f acc0 = {};
    v8f acc1 = {};

    // Prologue: stage chunk 0 into buffer 0.
    {
        const float* ps = Pb;
        async_b128(&sP[0][f0 * 4], ps + f0 * 4);
        async_b128(&sP[0][f1 * 4], ps + f1 * 4);
        async_b128(&sX[0][f0 * 4], xb + (size_t)xr0 * D_ + d0base + xc0);
        async_b128(&sX[0][f1 * 4], xb + (size_t)xr1 * D_ + d0base + xc1);
    }

    for (int c = 0; c < N_ / 32; ++c) {
        int buf = c & 1;
        wait_async0();          // my copies for chunk c are in LDS
        __syncthreads();        // everyone's copies are in LDS

        if (c + 1 < N_ / 32) {  // prefetch chunk c+1 into the other buffer
            int n0 = (c + 1) * 32;
            int nb = buf ^ 1;
            const float* ps = Pb + (size_t)n0 * K_;
            async_b128(&sP[nb][f0 * 4], ps + f0 * 4);
            async_b128(&sP[nb][f1 * 4], ps + f1 * 4);
            async_b128(&sX[nb][f0 * 4], xb + (size_t)(n0 + xr0) * D_ + d0base + xc0);
            async_b128(&sX[nb][f1 * 4], xb + (size_t)(n0 + xr1) * D_ + d0base + xc1);
        }

        #pragma unroll
        for (int s = 0; s < 8; ++s) {
            int nn = s * 4 + 2 * half;
            v2f af, b0, b1;
            af.x = sP[buf][(nn + 0) * 64 + k0 + l];
            af.y = sP[buf][(nn + 1) * 64 + k0 + l];
            b0.x = sX[buf][(nn + 0) * 64 + dg + l];
            b0.y = sX[buf][(nn + 1) * 64 + dg + l];
            b1.x = sX[buf][(nn + 0) * 64 + dg + 16 + l];
            b1.y = sX[buf][(nn + 1) * 64 + dg + 16 + l];
            acc0 = __builtin_amdgcn_wmma_f32_16x16x4_f32(
                false, af, false, b0, (short)0, acc0, false, false);
            acc1 = __builtin_amdgcn_wmma_f32_16x16x4_f32(
                false, af, false, b1, (short)0, acc1, false, false);
        }
        // No trailing barrier needed: reads of the buffer being prefetched next
        // completed (dscnt-waited into WMMAs) before this wave signals the next
        // iteration's barrier.
    }

    // C/D layout: VGPR r, lanes 0-15 -> M=r, N=l ; lanes 16-31 -> M=r+8, N=l.
    int mbase = half ? 8 : 0;
    float* ob = vlad + ((size_t)b * K_ + k0 + mbase) * D_ + d0base + dg + l;
    #pragma unroll
    for (int r = 0; r < 8; ++r) {
        ob[(size_t)r * D_]      = acc0[r];
        ob[(size_t)r * D_ + 16] = acc1[r];
    }
}

// ---------------------------------------------------------------------------
// Kernel 4: v = vlad - a_sum*c2^T, intra-normalize over D per (b,k),
// scatter to out[b, d*K + k], emit global-norm contribution ss/(ss+eps).
// ---------------------------------------------------------------------------
__global__ void __launch_bounds__(64) finalize1_k(const float* __restrict__ vlad,
                                                  const float* __restrict__ a_sum,
                                                  const float* __restrict__ c2,
                                                  float* __restrict__ out,
                                                  float* __restrict__ cbuf) {
    __shared__ float red[64];
    int b = blockIdx.x >> 6;
    int k = blockIdx.x & 63;
    int t = threadIdx.x;

    float as = a_sum[b * K_ + k];
    const float* vr = vlad + ((size_t)b * K_ + k) * D_;

    float v[8];
    float ss = 0.0f;
    #pragma unroll
    for (int i = 0; i < 8; ++i) {
        int d = t + i * 64;
        float val = vr[d] - as * c2[(size_t)d * K_ + k];  // clusters2[0,d,k]
        v[i] = val;
        ss += val * val;
    }
    red[t] = ss;
    __syncthreads();
    if (t < 32) {
        ss = red[t] + red[t + 32];
        #pragma unroll
        for (int off = 16; off > 0; off >>= 1) ss += __shfl_xor(ss, off, 32);
        if (t == 0) {
            red[0] = ss;
            cbuf[b * K_ + k] = ss / (ss + EPS_);  // sum of normalized squares
        }
    }
    __syncthreads();
    float scale = rsqrtf(red[0] + EPS_);
    float* ob = out + (size_t)b * (D_ * K_) + k;
    #pragma unroll
    for (int i = 0; i < 8; ++i) {
        int d = t + i * 64;
        ob[(size_t)d * K_] = v[i] * scale;
    }
}

// ---------------------------------------------------------------------------
// Kernel 5: per-batch global norm scale = rsqrt(sum_k cbuf[b,k] + eps).
// ---------------------------------------------------------------------------
__global__ void __launch_bounds__(32) gnorm_k(const float* __restrict__ cbuf,
                                              float* __restrict__ gscale) {
    int b = blockIdx.x;
    int lane = threadIdx.x;
    float s = cbuf[b * K_ + lane] + cbuf[b * K_ + lane + 32];
    #pragma unroll
    for (int off = 16; off > 0; off >>= 1) s += __shfl_xor(s, off, 32);
    if (lane == 0) gscale[b] = rsqrtf(s + EPS_);
}

// ---------------------------------------------------------------------------
// Kernel 6: out *= gscale[b], float4 vectorized.
// ---------------------------------------------------------------------------
__global__ void __launch_bounds__(256) scale_k(float* __restrict__ out,
                                               const float* __restrict__ gscale) {
    int idx = blockIdx.x * 256 + threadIdx.x;  // float4 index, 262144 total
    int b = idx >> 13;                         // 8192 float4 per batch
    float4 v = ((float4*)out)[idx];
    float s = gscale[b];
    v.x *= s; v.y *= s; v.z *= s; v.w *= s;
    ((float4*)out)[idx] = v;
}

// ---------------------------------------------------------------------------
extern "C" void kernel_launch(void* const* d_in, const int* in_sizes, int n_in,
                              void* d_out, int out_size, void* d_ws, size_t ws_size,
                              hipStream_t stream) {
    const float* x  = (const float*)d_in[0];   // [B,N,D]
    const float* bn = (const float*)d_in[1];   // [B*N,K]
    const float* c2 = (const float*)d_in[2];   // [1,D,K]
    float* out = (float*)d_out;                // [B, D*K]

    char* ws = (char*)d_ws;
    float* P      = (float*)(ws + 0);          //  8 MB  [B,N,K]
    float* vlad   = (float*)(ws + 8388608);    //  4 MB  [B,K,D]
    float* a_sum  = (float*)(ws + 12582912);   //  8 KB  [B,K]
    float* cbuf   = (float*)(ws + 12591104);   //  8 KB  [B,K]
    float* gscale = (float*)(ws + 12599296);   // 128 B  [B]

    softmax_k  <<<4096, 256, 0, stream>>>(bn, P);
    asum_k     <<<32,   256, 0, stream>>>(P, a_sum);
    vlad_gemm_k<<<256,  256, 0, stream>>>(P, x, vlad);
    finalize1_k<<<2048, 64,  0, stream>>>(vlad, a_sum, c2, out, cbuf);
    gnorm_k    <<<32,   32,  0, stream>>>(cbuf, gscale);
    scale_k    <<<1024, 256, 0, stream>>>(out, gscale);
}